// MotionCorrectBatch_57071525430231
// MI455X (gfx1250) — compile-verified
//
#include <hip/hip_runtime.h>
#include <math.h>

// ---------------------------------------------------------------------------
// MotionCorrectBatch for MI455X (gfx1250): phase correlation via DFT-as-GEMM
// on V_WMMA_F32_16X16X4_F32, partial inverse transform (only the 21x21
// fftshifted window is needed), integral-image NCC denominator, argmax +
// log-parabola subpixel, bilinear warp.
// ---------------------------------------------------------------------------

#define H 512
#define B 32
#define MS 10          // MS_H == MS_W == 10
#define WIN 21         // 2*MS+1
#define KSZ 492        // H - 2*MS
#define NPIX (H * H)   // 262144
#define PI_D 3.14159265358979323846

typedef __attribute__((ext_vector_type(2))) float v2f;
typedef __attribute__((ext_vector_type(8))) float v8f;

__device__ static inline v8f wmma4(v2f a, v2f b, v8f c) {
  // D = A(16x4) * B(4x16) + C(16x16), fp32
  return __builtin_amdgcn_wmma_f32_16x16x4_f32(
      /*neg_a=*/false, a, /*neg_b=*/false, b,
      /*c_mod=*/(short)0, c, /*reuse_a=*/false, /*reuse_b=*/false);
}

// --------------------------- table construction ----------------------------
// Wr/Wi: forward DFT matrix W[n][k] = exp(-2*pi*i*n*k/512) (symmetric).
// Er/Ei: partial inverse rows, E[j][k] = exp(+2*pi*i*k*y_j/512)/512,
//        y_j = (j-10) mod 512 for j<21, zero-padded to 32 rows.
__global__ void build_tables(float* __restrict__ Wr, float* __restrict__ Wi,
                             float* __restrict__ Er, float* __restrict__ Ei) {
  int idx = blockIdx.x * blockDim.x + threadIdx.x;
  if (idx < NPIX) {
    int n = idx >> 9, k = idx & 511;
    int t = (n * k) & 511;
    double ang = -2.0 * PI_D * (double)t / 512.0;
    Wr[idx] = (float)cos(ang);
    Wi[idx] = (float)sin(ang);
  }
  if (idx < 32 * H) {
    int j = idx >> 9, k = idx & 511;
    if (j < WIN) {
      int yj = (j - MS + H) & 511;
      int t = (k * yj) & 511;
      double ang = 2.0 * PI_D * (double)t / 512.0;
      Er[idx] = (float)(cos(ang) / 512.0);
      Ei[idx] = (float)(sin(ang) / 512.0);
    } else {
      Er[idx] = 0.f;
      Ei[idx] = 0.f;
    }
  }
}

// --------------------------- mean / var reduction --------------------------
__global__ void reduce_stats(const float* __restrict__ x, int n,
                             float* __restrict__ outMean,
                             float* __restrict__ outVar, int computeVar,
                             float eps) {
  const float* p = x + (size_t)blockIdx.x * n;
  double s = 0.0, sq = 0.0;
  for (int i = threadIdx.x; i < n; i += blockDim.x) {
    float v = p[i];
    s += (double)v;
    if (computeVar) sq += (double)v * (double)v;
  }
  __shared__ double sh[256];
  __shared__ double shq[256];
  sh[threadIdx.x] = s;
  shq[threadIdx.x] = sq;
  __syncthreads();
  for (int o = blockDim.x / 2; o > 0; o >>= 1) {
    if ((int)threadIdx.x < o) {
      sh[threadIdx.x] += sh[threadIdx.x + o];
      shq[threadIdx.x] += shq[threadIdx.x + o];
    }
    __syncthreads();
  }
  if (threadIdx.x == 0) {
    double mean = sh[0] / (double)n;
    outMean[blockIdx.x] = (float)mean;
    if (computeVar)
      outVar[blockIdx.x] = (float)(shq[0] - sh[0] * sh[0] / (double)n) + eps;
  }
}

// ----------------- pass 1: row DFT of real, mean-subtracted data -----------
// Y[row,k] = sum_n (X[row,n]-mu) * W[n,k]. One wave -> one 16x16 output tile,
// two accumulators (real/imag), K=512 in steps of 4.
__global__ void dft_rows_wmma(const float* __restrict__ X,
                              const float* __restrict__ means, int rowsPerFrame,
                              const float* __restrict__ Wr,
                              const float* __restrict__ Wi,
                              float* __restrict__ Yr, float* __restrict__ Yi) {
  const int lane = threadIdx.x;
  const int mL = lane & 15, hi = lane >> 4;
  const int row0 = (blockIdx.y * blockDim.y + threadIdx.y) * 16;
  const int col0 = blockIdx.x * 16;
  const float mu = means[row0 / rowsPerFrame];
  const float* xr = X + (size_t)(row0 + mL) * H;
  v8f accR = {};
  v8f accI = {};
#pragma unroll 4
  for (int k0 = 0; k0 < H; k0 += 4) {
    const int ka = k0 + 2 * hi;
    v2f a;
    a.x = xr[ka] - mu;
    a.y = xr[ka + 1] - mu;
    v2f br, bi;
    br.x = Wr[(size_t)ka * H + col0 + mL];
    br.y = Wr[(size_t)(ka + 1) * H + col0 + mL];
    bi.x = Wi[(size_t)ka * H + col0 + mL];
    bi.y = Wi[(size_t)(ka + 1) * H + col0 + mL];
    accR = wmma4(a, br, accR);
    accI = wmma4(a, bi, accI);
  }
#pragma unroll
  for (int v = 0; v < 8; ++v) {
    const int r = row0 + v + 8 * hi;
    Yr[(size_t)r * H + col0 + mL] = accR[v];
    Yi[(size_t)r * H + col0 + mL] = accI[v];
  }
}

// ------------- pass 2: column DFT (complex), optional conj(T) mult ---------
// Z[k1,k2] = sum_m W[k1,m]*Y[m,k2] (W symmetric). Complex = 4 real WMMA
// chains per K-step. Epilogue optionally multiplies by conj(T) elementwise.
__global__ void dft_cols_wmma(const float* __restrict__ Yr,
                              const float* __restrict__ Yi,
                              const float* __restrict__ Wr,
                              const float* __restrict__ Wi,
                              const float* __restrict__ Tr,
                              const float* __restrict__ Ti,
                              float* __restrict__ Zr, float* __restrict__ Zi,
                              int mulConj) {
  const int lane = threadIdx.x;
  const int mL = lane & 15, hi = lane >> 4;
  const int k20 = blockIdx.x * 16;
  const int k10 = (blockIdx.y * blockDim.y + threadIdx.y) * 16;
  const size_t foff = (size_t)blockIdx.z * NPIX;
  const float* ar_row = Wr + (size_t)(k10 + mL) * H;
  const float* ai_row = Wi + (size_t)(k10 + mL) * H;
  v8f zr = {};
  v8f zi = {};
#pragma unroll 4
  for (int k0 = 0; k0 < H; k0 += 4) {
    const int ka = k0 + 2 * hi;
    v2f ar, ai, nai;
    ar.x = ar_row[ka];
    ar.y = ar_row[ka + 1];
    ai.x = ai_row[ka];
    ai.y = ai_row[ka + 1];
    nai.x = -ai.x;            // F32 WMMA has no A/B NEG modifier -> VALU negate
    nai.y = -ai.y;
    v2f br, bi;
    br.x = Yr[foff + (size_t)ka * H + k20 + mL];
    br.y = Yr[foff + (size_t)(ka + 1) * H + k20 + mL];
    bi.x = Yi[foff + (size_t)ka * H + k20 + mL];
    bi.y = Yi[foff + (size_t)(ka + 1) * H + k20 + mL];
    zr = wmma4(ar, br, zr);
    zr = wmma4(nai, bi, zr);
    zi = wmma4(ar, bi, zi);
    zi = wmma4(ai, br, zi);
  }
#pragma unroll
  for (int v = 0; v < 8; ++v) {
    const size_t idx = (size_t)(k10 + v + 8 * hi) * H + k20 + mL;
    float a = zr[v], b = zi[v];
    if (mulConj) {
      float tr = Tr[idx], ti = Ti[idx];
      Zr[foff + idx] = a * tr + b * ti;   // Re(Z * conj(T))
      Zi[foff + idx] = b * tr - a * ti;   // Im(Z * conj(T))
    } else {
      Zr[foff + idx] = a;
      Zi[foff + idx] = b;
    }
  }
}

// ---------- pass 3: partial inverse over k1 (only 21 output rows) ----------
// Q[j,k2] = sum_k1 E[j,k1]*P[k1,k2], complex, M=32 (21 valid, zero-padded).
__global__ void inv_cols_wmma(const float* __restrict__ Pr,
                              const float* __restrict__ Pi,
                              const float* __restrict__ Er,
                              const float* __restrict__ Ei,
                              float* __restrict__ Qr, float* __restrict__ Qi) {
  const int lane = threadIdx.x;
  const int mL = lane & 15, hi = lane >> 4;
  const int k20 = blockIdx.x * 16;
  const int j0 = threadIdx.y * 16;
  const size_t foff = (size_t)blockIdx.z * NPIX;
  const size_t qoff = (size_t)blockIdx.z * (32 * H);
  const float* er_row = Er + (size_t)(j0 + mL) * H;
  const float* ei_row = Ei + (size_t)(j0 + mL) * H;
  v8f qr = {};
  v8f qi = {};
#pragma unroll 4
  for (int k0 = 0; k0 < H; k0 += 4) {
    const int ka = k0 + 2 * hi;
    v2f ar, ai, nai;
    ar.x = er_row[ka];
    ar.y = er_row[ka + 1];
    ai.x = ei_row[ka];
    ai.y = ei_row[ka + 1];
    nai.x = -ai.x;
    nai.y = -ai.y;
    v2f br, bi;
    br.x = Pr[foff + (size_t)ka * H + k20 + mL];
    br.y = Pr[foff + (size_t)(ka + 1) * H + k20 + mL];
    bi.x = Pi[foff + (size_t)ka * H + k20 + mL];
    bi.y = Pi[foff + (size_t)(ka + 1) * H + k20 + mL];
    qr = wmma4(ar, br, qr);
    qr = wmma4(nai, bi, qr);
    qi = wmma4(ar, bi, qi);
    qi = wmma4(ai, br, qi);
  }
#pragma unroll
  for (int v = 0; v < 8; ++v) {
    const size_t idx = qoff + (size_t)(j0 + v + 8 * hi) * H + k20 + mL;
    Qr[idx] = qr[v];
    Qi[idx] = qi[v];
  }
}

// ------------- final 21x21 projection + magnitude (tiny, VALU) -------------
__global__ void nom_kernel(const float* __restrict__ Qr,
                           const float* __restrict__ Qi,
                           const float* __restrict__ Er,
                           const float* __restrict__ Ei,
                           float* __restrict__ nom) {
  const int f = blockIdx.x;
  const int t = threadIdx.x;
  if (t >= WIN * WIN) return;
  const int jy = t / WIN, jx = t % WIN;
  const float* qr = Qr + (size_t)f * (32 * H) + (size_t)jy * H;
  const float* qi = Qi + (size_t)f * (32 * H) + (size_t)jy * H;
  const float* er = Er + (size_t)jx * H;
  const float* ei = Ei + (size_t)jx * H;
  float re = 0.f, im = 0.f;
  for (int k = 0; k < H; ++k) {
    float a = qr[k], b = qi[k], c = er[k], d = ei[k];
    re += a * c - b * d;
    im += a * d + b * c;
  }
  nom[f * (WIN * WIN) + t] = sqrtf(re * re + im * im);
}

// --------------------- integral images (sum, sum of squares) ---------------
__global__ void row_prefix(const float* __restrict__ X, float* __restrict__ Is,
                           float* __restrict__ Iq) {
  const int r = blockIdx.x * blockDim.x + threadIdx.x;
  if (r >= B * H) return;
  const float* x = X + (size_t)r * H;
  float* s = Is + (size_t)r * H;
  float* q = Iq + (size_t)r * H;
  float a = 0.f, b = 0.f;
  for (int i = 0; i < H; ++i) {
    float v = x[i];
    a += v;
    b += v * v;
    s[i] = a;
    q[i] = b;
  }
}

__global__ void col_prefix(float* __restrict__ Is, float* __restrict__ Iq) {
  const int c = blockIdx.x * blockDim.x + threadIdx.x;
  if (c >= B * H) return;
  const int f = c >> 9, x = c & 511;
  float* s = Is + (size_t)f * NPIX + x;
  float* q = Iq + (size_t)f * NPIX + x;
  float a = s[0], b = q[0];
  for (int y = 1; y < H; ++y) {
    a += s[(size_t)y * H];
    s[(size_t)y * H] = a;
    b += q[(size_t)y * H];
    q[(size_t)y * H] = b;
  }
}

// ----------------- NCC, argmax, log-parabola subpixel shifts ---------------
__global__ void finalize_shifts(const float* __restrict__ nom,
                                const float* __restrict__ Is,
                                const float* __restrict__ Iq,
                                const float* __restrict__ tvar,
                                float* __restrict__ shx_out,
                                float* __restrict__ shy_out) {
  const int f = blockIdx.x;
  const int t = threadIdx.x;
  __shared__ float sncc[WIN * WIN];
  __shared__ float sval[512];
  __shared__ int sidx[512];
  const float* I = Is + (size_t)f * NPIX;
  const float* Q = Iq + (size_t)f * NPIX;
  float v = -1.0f;  // ncc >= 0 (or NaN->0), safe pad
  if (t < WIN * WIN) {
    const int i = t / WIN, j = t % WIN;
    const int y2 = i + KSZ - 1, x2 = j + KSZ - 1;
    float S = I[y2 * H + x2];
    float Sq = Q[y2 * H + x2];
    if (i > 0) { S -= I[(i - 1) * H + x2]; Sq -= Q[(i - 1) * H + x2]; }
    if (j > 0) { S -= I[y2 * H + (j - 1)]; Sq -= Q[y2 * H + (j - 1)]; }
    if (i > 0 && j > 0) { S += I[(i - 1) * H + (j - 1)]; Sq += Q[(i - 1) * H + (j - 1)]; }
    const float KW = (float)KSZ * (float)KSZ;
    float m1 = S / KW;
    float m2 = Sq / KW;
    float var = m2 - (m1 * m1) / KW + 1e-8f;
    if (var < 0.f) var = 0.f;
    float den = sqrtf(tvar[0] * var);
    float n = nom[f * (WIN * WIN) + t] / den;
    if (n != n) n = 0.f;  // NaN -> 0 (matches reference; Inf passes through)
    sncc[t] = n;
    v = n;
  }
  sval[t] = v;
  sidx[t] = t;
  __syncthreads();
  for (int o = 256; o > 0; o >>= 1) {
    if (t < o) {
      float v2 = sval[t + o];
      int i2 = sidx[t + o];
      if (v2 > sval[t] || (v2 == sval[t] && i2 < sidx[t])) {
        sval[t] = v2;
        sidx[t] = i2;
      }
    }
    __syncthreads();
  }
  if (t == 0) {
    const int am = sidx[0];
    const int sx = am / WIN, sy = am % WIN;
    // JAX dynamic-index semantics: negative wraps, overflow clamps
    int xm = sx - 1; if (xm < 0) xm += WIN;
    int xp = sx + 1; if (xp > WIN - 1) xp = WIN - 1;
    int ym = sy - 1; if (ym < 0) ym += WIN;
    int yp = sy + 1; if (yp > WIN - 1) yp = WIN - 1;
    const float lxm1 = logf(sncc[xm * WIN + sy]);
    const float lxp1 = logf(sncc[xp * WIN + sy]);
    const float lym1 = logf(sncc[sx * WIN + ym]);
    const float lyp1 = logf(sncc[sx * WIN + yp]);
    const float lc4 = 4.f * logf(sncc[sx * WIN + sy]);
    float sh_x = -(float)(sx - MS) - (lxm1 - lxp1) / (2.f * lxm1 - lc4 + 2.f * lxp1);
    float sh_y = -(float)(sy - MS) - (lym1 - lyp1) / (2.f * lym1 - lc4 + 2.f * lyp1);
    shx_out[f] = sh_x;
    shy_out[f] = sh_y;
  }
}

// --------------------- bilinear warp + transposed store --------------------
__global__ void warp_kernel(const float* __restrict__ X,
                            const float* __restrict__ shx,
                            const float* __restrict__ shy,
                            float* __restrict__ out) {
  const int b = blockIdx.y;
  const int p = blockIdx.x * blockDim.x + threadIdx.x;
  const int y = p >> 9, x = p & 511;
  const float dy = shx[b], dx = shy[b];
  const float yq = (float)y - dy, xq = (float)x - dx;
  const float y0f = floorf(yq), x0f = floorf(xq);
  const float wy = yq - y0f, wx = xq - x0f;
  const int y0 = (int)y0f, x0 = (int)x0f;
  const float* img = X + (size_t)b * NPIX;
  auto samp = [&](int yi, int xi) -> float {
    bool ok = (yi >= 0) && (yi < H) && (xi >= 0) && (xi < H);
    int yc = yi < 0 ? 0 : (yi > H - 1 ? H - 1 : yi);
    int xc = xi < 0 ? 0 : (xi > H - 1 ? H - 1 : xi);
    float vv = img[(size_t)yc * H + xc];
    return ok ? vv : 0.f;
  };
  const float v00 = samp(y0, x0);
  const float v01 = samp(y0, x0 + 1);
  const float v10 = samp(y0 + 1, x0);
  const float v11 = samp(y0 + 1, x0 + 1);
  const float r = v00 * (1.f - wy) * (1.f - wx) + v01 * (1.f - wy) * wx +
                  v10 * wy * (1.f - wx) + v11 * wy * wx;
  // transpose(0,2,1).reshape(B, H*W): out[b, x*H + y] = warped[b, y, x]
  out[(size_t)b * NPIX + (size_t)x * H + y] = r;
}

// ---------------------------------------------------------------------------
extern "C" void kernel_launch(void* const* d_in, const int* in_sizes, int n_in,
                              void* d_out, int out_size, void* d_ws,
                              size_t ws_size, hipStream_t stream) {
  (void)in_sizes; (void)n_in; (void)out_size; (void)ws_size;
  const float* fr = (const float*)d_in[0];     // (1,32,512,512,1) f32
  const float* tmpl = (const float*)d_in[1];   // (512,512) f32
  float* out = (float*)d_out;

  float* w = (float*)d_ws;
  size_t off = 0;
  auto alloc = [&](size_t n) { float* p = w + off; off += n; return p; };
  float* Wr = alloc(NPIX);
  float* Wi = alloc(NPIX);
  float* Er = alloc(32 * H);
  float* Ei = alloc(32 * H);
  float* Tr = alloc(NPIX);
  float* Ti = alloc(NPIX);
  float* Yr = alloc((size_t)B * NPIX);
  float* Yi = alloc((size_t)B * NPIX);
  float* Pr = alloc((size_t)B * NPIX);
  float* Pi = alloc((size_t)B * NPIX);
  float* Qr = alloc((size_t)B * 32 * H);
  float* Qi = alloc((size_t)B * 32 * H);
  float* nomb = alloc(B * WIN * WIN);
  float* means = alloc(B);
  float* tmean = alloc(1);
  float* tvar = alloc(1);
  float* shx = alloc(B);
  float* shy = alloc(B);
  // Integral images reuse Yr/Yi, which are dead after the frame column-DFT.
  float* Is = Yr;
  float* Iq = Yi;

  // DFT / partial-inverse matrices
  build_tables<<<(NPIX + 255) / 256, 256, 0, stream>>>(Wr, Wi, Er, Ei);
  // per-frame means; template mean + variance
  reduce_stats<<<B, 256, 0, stream>>>(fr, NPIX, means, nullptr, 0, 0.f);
  reduce_stats<<<1, 256, 0, stream>>>(tmpl, NPIX, tmean, tvar, 1, 1e-8f);
  // template 2D FFT (rows then cols), staged through Yr/Yi
  dft_rows_wmma<<<dim3(32, 8), dim3(32, 4), 0, stream>>>(tmpl, tmean, H, Wr, Wi,
                                                         Yr, Yi);
  dft_cols_wmma<<<dim3(32, 8, 1), dim3(32, 4), 0, stream>>>(
      Yr, Yi, Wr, Wi, nullptr, nullptr, Tr, Ti, 0);
  // frames: row FFT, then column FFT fused with conj(T) product
  dft_rows_wmma<<<dim3(32, 256), dim3(32, 4), 0, stream>>>(fr, means, H, Wr, Wi,
                                                           Yr, Yi);
  dft_cols_wmma<<<dim3(32, 8, B), dim3(32, 4), 0, stream>>>(Yr, Yi, Wr, Wi, Tr,
                                                            Ti, Pr, Pi, 1);
  // partial inverse over k1 (21 needed rows), then 21x21 projection + |.|
  inv_cols_wmma<<<dim3(32, 1, B), dim3(32, 2), 0, stream>>>(Pr, Pi, Er, Ei, Qr,
                                                            Qi);
  nom_kernel<<<B, 448, 0, stream>>>(Qr, Qi, Er, Ei, nomb);
  // windowed-variance denominator via integral images (overwrites Yr/Yi)
  row_prefix<<<(B * H) / 256, 256, 0, stream>>>(fr, Is, Iq);
  col_prefix<<<(B * H) / 256, 256, 0, stream>>>(Is, Iq);
  // NCC, argmax, subpixel shifts
  finalize_shifts<<<B, 512, 0, stream>>>(nomb, Is, Iq, tvar, shx, shy);
  // bilinear warp + transposed flatten
  warp_kernel<<<dim3(NPIX / 256, B), 256, 0, stream>>>(fr, shx, shy, out);
}